// MalwareGNN_8718783610906
// MI455X (gfx1250) — compile-verified
//
#include <hip/hip_runtime.h>
#include <hip/hip_bf16.h>

// GCN forward for MI455X (gfx1250, wave32).
// Dense transforms use V_WMMA_F32_16X16X4_F32 (f32 A/B/C -> full precision).
// Sparse aggregation: wave-per-edge gather + global_atomic_add_f32 scatter.

#define FIN   128
#define HIDD  64
#define NCLS  10

typedef float v2f __attribute__((ext_vector_type(2)));
typedef float v8f __attribute__((ext_vector_type(8)));

#if defined(__has_builtin)
#if __has_builtin(__builtin_amdgcn_wmma_f32_16x16x4_f32)
#define HAVE_WMMA_F32X4 1
#endif
#endif

// ---------------------------------------------------------------- utilities
__global__ void k_fill(float* __restrict__ p, float v, int n) {
    int i = blockIdx.x * blockDim.x + threadIdx.x;
    if (i < n) p[i] = v;
}

__global__ void k_deg(const long long* __restrict__ ei, float* __restrict__ deg,
                      long long E) {
    long long e = (long long)blockIdx.x * blockDim.x + threadIdx.x;
    if (e < E) atomicAdd(&deg[ei[E + e]], 1.0f);   // dst = ei[1][e]
}

__global__ void k_rsqrt_inplace(float* __restrict__ p, int n) {
    int i = blockIdx.x * blockDim.x + threadIdx.x;
    if (i < n) p[i] = rsqrtf(p[i]);
}

// ------------------------------------------------------------- dense GEMM
// Out[N x 64] = In[N x K] @ W[K x 64].  Grid: N/16 blocks of 128 threads.
// Wave w owns 16-col tile w; block owns a 16-row slab. No divergence ->
// EXEC all ones as required by WMMA.
__global__ void k_gemm(const float* __restrict__ In, const float* __restrict__ W,
                       float* __restrict__ Out, int K) {
    const int lane = threadIdx.x & 31;
    const int wave = threadIdx.x >> 5;       // 0..3 -> column tile
    const int half = lane >> 4;              // 0: lanes 0-15, 1: lanes 16-31
    const int l    = lane & 15;
    const int row0 = blockIdx.x << 4;
    const int col  = (wave << 4) + l;

    v8f acc = {};
#if defined(HAVE_WMMA_F32X4)
    // A frag (16x4 f32): lanes 0-15 hold K=0,1 ; lanes 16-31 hold K=2,3.
    // B frag (4x16 f32): mirrored; C/D per documented 16x16 f32 layout.
    for (int k0 = 0; k0 < K; k0 += 4) {
        const int ka = k0 + (half << 1);
        v2f a, b;
        a.x = In[(long long)(row0 + l) * K + ka];
        a.y = In[(long long)(row0 + l) * K + ka + 1];
        b.x = W[ka * HIDD + col];
        b.y = W[(ka + 1) * HIDD + col];
        acc = __builtin_amdgcn_wmma_f32_16x16x4_f32(
            /*neg_a=*/false, a, /*neg_b=*/false, b,
            /*c_mod=*/(short)0, acc, /*reuse_a=*/false, /*reuse_b=*/false);
    }
#else
    // Scalar fallback with identical output mapping.
    for (int v = 0; v < 8; ++v) {
        const int row = row0 + v + (half << 3);
        float s = 0.0f;
        for (int k = 0; k < K; ++k)
            s = fmaf(In[(long long)row * K + k], W[k * HIDD + col], s);
        acc[v] = s;
    }
#endif
    #pragma unroll
    for (int v = 0; v < 8; ++v) {
        const int row = row0 + v + (half << 3);
        Out[(long long)row * HIDD + col] = acc[v];
    }
}

// ------------------------------------------------------- sparse aggregation
// Self-loop contribution: agg[i] = dinv[i]^2 * h[i]
__global__ void k_selfloop(const float* __restrict__ h, const float* __restrict__ dinv,
                           float* __restrict__ agg, int n64) {
    int idx = blockIdx.x * blockDim.x + threadIdx.x;
    if (idx < n64) {
        float di = dinv[idx >> 6];
        agg[idx] = di * di * h[idx];
    }
}

// One wave per edge: lane f covers features f and f+32 (coalesced gather,
// two global_atomic_add_f32 scatters).
__global__ void k_edge_agg(const float* __restrict__ h, float* __restrict__ agg,
                           const long long* __restrict__ ei,
                           const float* __restrict__ dinv, long long E) {
    const int lane = threadIdx.x & 31;
    const long long e = (long long)blockIdx.x * (blockDim.x >> 5) + (threadIdx.x >> 5);
    if (e >= E) return;
    const long long s = ei[e];       // src
    const long long d = ei[E + e];   // dst
    const float nrm = dinv[s] * dinv[d];
    const float* __restrict__ hs = h + s * HIDD;
    float* __restrict__ od = agg + d * HIDD;
    atomicAdd(&od[lane],      nrm * hs[lane]);
    atomicAdd(&od[lane + 32], nrm * hs[lane + 32]);
}

__global__ void k_bias_relu(float* __restrict__ h, const float* __restrict__ b, int n64) {
    int idx = blockIdx.x * blockDim.x + threadIdx.x;
    if (idx < n64) h[idx] = fmaxf(h[idx] + b[idx & 63], 0.0f);
}

// ----------------------------------------------------------------- pooling
__global__ void k_pool(const float* __restrict__ h, const long long* __restrict__ batch,
                       float* __restrict__ pooled, float* __restrict__ cnt, int n64) {
    int idx = blockIdx.x * blockDim.x + threadIdx.x;
    if (idx < n64) {
        const int i = idx >> 6, f = idx & 63;
        const long long g = batch[i];
        atomicAdd(&pooled[g * HIDD + f], h[idx]);
        if (f == 0) atomicAdd(&cnt[g], 1.0f);
    }
}

__global__ void k_classify(const float* __restrict__ pooled, const float* __restrict__ cnt,
                           const float* __restrict__ Wc, const float* __restrict__ bc,
                           float* __restrict__ out, int nout) {
    int t = blockIdx.x * blockDim.x + threadIdx.x;
    if (t >= nout) return;
    const int g = t / NCLS, c = t % NCLS;
    const float inv = 1.0f / fmaxf(cnt[g], 1.0f);
    float s = bc[c];
    for (int k = 0; k < HIDD; ++k)
        s = fmaf(pooled[g * HIDD + k] * inv, Wc[k * NCLS + c], s);
    out[t] = s;
}

// ------------------------------------------------------------------ driver
extern "C" void kernel_launch(void* const* d_in, const int* in_sizes, int n_in,
                              void* d_out, int out_size, void* d_ws, size_t ws_size,
                              hipStream_t stream) {
    const float*     x     = (const float*)d_in[0];
    const long long* ei    = (const long long*)d_in[1];
    const long long* batch = (const long long*)d_in[2];
    const float* W1 = (const float*)d_in[3];  const float* b1 = (const float*)d_in[4];
    const float* W2 = (const float*)d_in[5];  const float* b2 = (const float*)d_in[6];
    const float* W3 = (const float*)d_in[7];  const float* b3 = (const float*)d_in[8];
    const float* Wc = (const float*)d_in[9];  const float* bc = (const float*)d_in[10];
    float* out = (float*)d_out;

    const int       N   = in_sizes[0] / FIN;          // 100000
    const long long E   = (long long)in_sizes[1] / 2; // 3200000
    const int       G   = out_size / NCLS;            // 256
    const int       n64 = N * HIDD;

    // workspace layout (floats)
    float* dinv   = (float*)d_ws;            // N
    float* hA     = dinv + N;                // N*64  (transform result)
    float* hB     = hA + n64;                // N*64  (aggregation / layer output)
    float* pooled = hB + n64;                // G*64
    float* cnt    = pooled + G * HIDD;       // G

    const int T = 256;
    // --- degree -> dinv ---
    k_fill<<<(N + T - 1) / T, T, 0, stream>>>(dinv, 1.0f, N);   // self loop
    k_deg<<<(int)((E + T - 1) / T), T, 0, stream>>>(ei, dinv, E);
    k_rsqrt_inplace<<<(N + T - 1) / T, T, 0, stream>>>(dinv, N);

    const int gemmBlocks = N / 16;           // 6250, exact
    const int edgeBlocks = (int)((E + 7) / 8);

    // --- layer 1 ---
    k_gemm<<<gemmBlocks, 128, 0, stream>>>(x, W1, hA, FIN);
    k_selfloop<<<(n64 + T - 1) / T, T, 0, stream>>>(hA, dinv, hB, n64);
    k_edge_agg<<<edgeBlocks, T, 0, stream>>>(hA, hB, ei, dinv, E);
    k_bias_relu<<<(n64 + T - 1) / T, T, 0, stream>>>(hB, b1, n64);

    // --- layer 2 ---
    k_gemm<<<gemmBlocks, 128, 0, stream>>>(hB, W2, hA, HIDD);
    k_selfloop<<<(n64 + T - 1) / T, T, 0, stream>>>(hA, dinv, hB, n64);
    k_edge_agg<<<edgeBlocks, T, 0, stream>>>(hA, hB, ei, dinv, E);
    k_bias_relu<<<(n64 + T - 1) / T, T, 0, stream>>>(hB, b2, n64);

    // --- layer 3 ---
    k_gemm<<<gemmBlocks, 128, 0, stream>>>(hB, W3, hA, HIDD);
    k_selfloop<<<(n64 + T - 1) / T, T, 0, stream>>>(hA, dinv, hB, n64);
    k_edge_agg<<<edgeBlocks, T, 0, stream>>>(hA, hB, ei, dinv, E);
    k_bias_relu<<<(n64 + T - 1) / T, T, 0, stream>>>(hB, b3, n64);

    // --- global mean pool + classifier ---
    k_fill<<<(G * HIDD + G + T - 1) / T, T, 0, stream>>>(pooled, 0.0f, G * HIDD + G);
    k_pool<<<(n64 + T - 1) / T, T, 0, stream>>>(hB, batch, pooled, cnt, n64);
    k_classify<<<(G * NCLS + 63) / 64, 64, 0, stream>>>(pooled, cnt, Wc, bc, out, G * NCLS);
}